// TrainableSeparateGroupPositionalEncoding_45260365365582
// MI455X (gfx1250) — compile-verified
//
#include <hip/hip_runtime.h>

// out[t] = x[t] + (mask[t] ? pe_table[prefix_in_batch(t)] : 0)
// prefix_in_batch(t) = # masked tokens before t inside t's 4096-token batch.
//
// Two-kernel pipeline:
//   A) per-batch wave32 ballot/shuffle scan -> u16 pe index per token in d_ws
//   B) 8192-block streaming gather-add: NT B128 x/out streams, RT pe gather,
//      async-to-LDS staging of the per-block index slice.

typedef float v4f __attribute__((ext_vector_type(4)));
typedef unsigned short u16;

#define SEQ          4096            // tokens per batch segment
#define DVEC         64              // 256 floats = 64 float4 per row
#define SCAN_THREADS 1024            // 32 waves
#define SCAN_WAVES   (SCAN_THREADS / 32)
#define ADD_THREADS  256             // 8 waves
#define ADD_WAVES    (ADD_THREADS / 32)
#define ROWS_PER_BLK 64

// ---------------------------------------------------------------------------
// Kernel A: per-batch exclusive prefix sum of the mask -> u16 indices in ws.
// One 1024-thread block per 4096-token batch; thread owns 4 consecutive tokens.
// ---------------------------------------------------------------------------
__global__ __launch_bounds__(SCAN_THREADS, 1)
void scan_kernel(const int* __restrict__ maskg, u16* __restrict__ ws)
{
    __shared__ int wave_tot[SCAN_WAVES];
    __shared__ int wave_excl[SCAN_WAVES];

    const int tid  = threadIdx.x;
    const int lane = tid & 31;
    const int wave = tid >> 5;
    const size_t base = (size_t)blockIdx.x * SEQ;

    const int t0 = tid * 4;
    int4 mv = *(const int4*)(maskg + base + t0);
    const int m0 = (mv.x != 0), m1 = (mv.y != 0), m2 = (mv.z != 0), m3 = (mv.w != 0);
    const int s  = m0 + m1 + m2 + m3;

    // inclusive wave32 scan of per-thread sums (5 shuffle steps)
    int v = s;
    #pragma unroll
    for (int off = 1; off < 32; off <<= 1) {
        int t = __shfl_up(v, off, 32);
        if (lane >= off) v += t;
    }
    if (lane == 31) wave_tot[wave] = v;
    __syncthreads();

    if (wave == 0) {                       // scan the 32 wave totals
        int w  = wave_tot[lane];
        int wv = w;
        #pragma unroll
        for (int off = 1; off < 32; off <<= 1) {
            int t = __shfl_up(wv, off, 32);
            if (lane >= off) wv += t;
        }
        wave_excl[lane] = wv - w;
    }
    __syncthreads();

    int p = wave_excl[wave] + (v - s);     // exclusive prefix of this thread's chunk
    ushort4 q;
    q.x = m0 ? (u16)p : (u16)0xFFFF; p += m0;
    q.y = m1 ? (u16)p : (u16)0xFFFF; p += m1;
    q.z = m2 ? (u16)p : (u16)0xFFFF; p += m2;
    q.w = m3 ? (u16)p : (u16)0xFFFF;
    *(ushort4*)(ws + base + t0) = q;       // coalesced 8 B per thread
}

// ---------------------------------------------------------------------------
// Kernel B: out = x (+ pe[idx]).  64 rows per 256-thread block (8 waves).
// Index slice staged to LDS via CDNA5 async-to-LDS (ASYNCcnt-tracked).
// ---------------------------------------------------------------------------
__global__ __launch_bounds__(ADD_THREADS)
void add_kernel(const float* __restrict__ xg,
                const float* __restrict__ peg,
                const u16*   __restrict__ ws,
                float*       __restrict__ outg)
{
    // ONLY shared object in this kernel => LDS byte offset 0, so the async
    // VDST VGPR (LDS address taken from it) is just the array-relative offset.
    __shared__ u16 idx_lds[ROWS_PER_BLK];  // 128 B

    const int tid  = threadIdx.x;
    const int lane = tid & 31;
    const int wave = tid >> 5;
    const size_t row0 = (size_t)blockIdx.x * ROWS_PER_BLK;

    // Stage 64 u16 indices (128 B) into LDS: 8 lanes x async B128, GVS mode.
    if (tid < 8) {
        unsigned lds_off = tid * 16;       // VDST: LDS byte address
        unsigned g_off   = tid * 16;       // VADDR: byte offset from SADDR
        asm volatile(
            "global_load_async_to_lds_b128 %0, %1, %2"
            :
            : "v"(lds_off), "v"(g_off), "s"(ws + row0)
            : "memory");
    }
    asm volatile("s_wait_asynccnt 0x0" ::: "memory");
    __syncthreads();

    const v4f* __restrict__ xb  = (const v4f*)xg;
    const v4f* __restrict__ peb = (const v4f*)peg;
    v4f*       __restrict__ ob  = (v4f*)outg;

    for (int i = wave; i < ROWS_PER_BLK; i += ADD_WAVES) {
        const unsigned pidx = idx_lds[i];           // LDS broadcast read
        const size_t   ro   = (row0 + (size_t)i) * DVEC;

        if (i + ADD_WAVES < ROWS_PER_BLK)           // global_prefetch_b8
            __builtin_prefetch(&xb[ro + (size_t)ADD_WAVES * DVEC + lane], 0, 0);

        v4f a = __builtin_nontemporal_load(&xb[ro + lane]);
        v4f b = __builtin_nontemporal_load(&xb[ro + lane + 32]);

        if (pidx != 0xFFFFu) {
            const size_t po = (size_t)pidx * DVEC;
            a += peb[po + lane];                    // RT: pe_table stays in L2
            b += peb[po + lane + 32];
        }

        __builtin_nontemporal_store(a, &ob[ro + lane]);
        __builtin_nontemporal_store(b, &ob[ro + lane + 32]);
    }
}

// ---------------------------------------------------------------------------
// Fallback: fused single-kernel version (used only if d_ws is too small).
// ---------------------------------------------------------------------------
__global__ __launch_bounds__(SCAN_THREADS, 1)
void pe_add_fused_kernel(const float* __restrict__ xg,
                         const int*   __restrict__ maskg,
                         const float* __restrict__ peg,
                         float*       __restrict__ outg)
{
    __shared__ u16 pidx_lds[SEQ];
    __shared__ int wave_tot[SCAN_WAVES];
    __shared__ int wave_excl[SCAN_WAVES];

    const int tid  = threadIdx.x;
    const int lane = tid & 31;
    const int wave = tid >> 5;
    const size_t base = (size_t)blockIdx.x * SEQ;

    const int t0 = tid * 4;
    int4 mv = *(const int4*)(maskg + base + t0);
    const int m0 = (mv.x != 0), m1 = (mv.y != 0), m2 = (mv.z != 0), m3 = (mv.w != 0);
    const int s  = m0 + m1 + m2 + m3;

    int v = s;
    #pragma unroll
    for (int off = 1; off < 32; off <<= 1) {
        int t = __shfl_up(v, off, 32);
        if (lane >= off) v += t;
    }
    if (lane == 31) wave_tot[wave] = v;
    __syncthreads();
    if (wave == 0) {
        int w  = wave_tot[lane];
        int wv = w;
        #pragma unroll
        for (int off = 1; off < 32; off <<= 1) {
            int t = __shfl_up(wv, off, 32);
            if (lane >= off) wv += t;
        }
        wave_excl[lane] = wv - w;
    }
    __syncthreads();

    int p = wave_excl[wave] + (v - s);
    pidx_lds[t0 + 0] = m0 ? (u16)p : (u16)0xFFFF; p += m0;
    pidx_lds[t0 + 1] = m1 ? (u16)p : (u16)0xFFFF; p += m1;
    pidx_lds[t0 + 2] = m2 ? (u16)p : (u16)0xFFFF; p += m2;
    pidx_lds[t0 + 3] = m3 ? (u16)p : (u16)0xFFFF;
    __syncthreads();

    const v4f* __restrict__ xb  = (const v4f*)xg;
    const v4f* __restrict__ peb = (const v4f*)peg;
    v4f*       __restrict__ ob  = (v4f*)outg;

    for (int r = wave; r < SEQ; r += SCAN_WAVES) {
        const unsigned pidx = pidx_lds[r];
        const size_t   ro   = (base + (size_t)r) * DVEC;
        v4f a = __builtin_nontemporal_load(&xb[ro + lane]);
        v4f b = __builtin_nontemporal_load(&xb[ro + lane + 32]);
        if (pidx != 0xFFFFu) {
            const size_t po = (size_t)pidx * DVEC;
            a += peb[po + lane];
            b += peb[po + lane + 32];
        }
        __builtin_nontemporal_store(a, &ob[ro + lane]);
        __builtin_nontemporal_store(b, &ob[ro + lane + 32]);
    }
}

extern "C" void kernel_launch(void* const* d_in, const int* in_sizes, int n_in,
                              void* d_out, int out_size, void* d_ws, size_t ws_size,
                              hipStream_t stream) {
    // setup_inputs() order: x, local_indices, group_mask, batch_indicator, pe_table
    const float* x    = (const float*)d_in[0];
    const int*   mask = (const int*)d_in[2];
    const float* pe   = (const float*)d_in[4];
    float*       out  = (float*)d_out;
    (void)n_in; (void)out_size;

    const int n_tokens = in_sizes[1];            // local_indices count == N_TOKENS
    const int batches  = n_tokens / SEQ;

    if (ws_size >= (size_t)n_tokens * sizeof(u16)) {
        u16* ws = (u16*)d_ws;
        scan_kernel<<<batches, SCAN_THREADS, 0, stream>>>(mask, ws);
        add_kernel<<<n_tokens / ROWS_PER_BLK, ADD_THREADS, 0, stream>>>(x, pe, ws, out);
    } else {
        pe_add_fused_kernel<<<batches, SCAN_THREADS, 0, stream>>>(x, mask, pe, out);
    }
}